// update_v_40638980555086
// MI455X (gfx1250) — compile-verified
//
#include <hip/hip_runtime.h>
#include <hip/hip_bf16.h>
#include <math.h>

typedef __attribute__((ext_vector_type(2))) float v2f;
typedef __attribute__((ext_vector_type(8))) float v8f;

#define HID 128
#define OEC 256

// ---------------------------------------------------------------------------
// zero a float buffer (ws is poisoned 0xAA; we must clear accumulation dests)
// ---------------------------------------------------------------------------
__global__ void zero_kernel(float* __restrict__ p, long long n) {
  long long i = (long long)blockIdx.x * blockDim.x + threadIdx.x;
  long long stride = (long long)gridDim.x * blockDim.x;
  for (; i < n; i += stride) p[i] = 0.0f;
}

// ---------------------------------------------------------------------------
// v0[i[e], :] += e2[e, :]  -- float4 per thread (b128 load + 4 f32 atomics)
// total = M * 32 threads
// ---------------------------------------------------------------------------
__global__ void scatter_rows_kernel(const float* __restrict__ src,
                                    const int* __restrict__ idx,
                                    float* __restrict__ dst, long long total) {
  long long gid = (long long)blockIdx.x * blockDim.x + threadIdx.x;
  if (gid >= total) return;
  const int e = (int)(gid >> 5);
  const int t = (int)((gid & 31) << 2);
  const float4 x = *(const float4*)(src + (long long)e * HID + t);
  float* o = dst + (long long)idx[e] * HID + t;
  atomicAdd(o + 0, x.x);
  atomicAdd(o + 1, x.y);
  atomicAdd(o + 2, x.z);
  atomicAdd(o + 3, x.w);
}

// ---------------------------------------------------------------------------
// Templated fp32 WMMA GEMM: out[rows, cols] = act(A') @ W^T (+bias, +relu)
//   A'[r, k] = k < K1 ? A[r*K1+k] : A2[r*K2+(k-K1)]    (concat, lin_v)
//   W is [cols, K1+K2] row-major (PyTorch Linear layout)
// K1 % 4 == 0 required; main loop is guard-free with b64 A/B loads.
// One wave owns one 16x16 tile (V_WMMA_F32_16X16X4_F32, K step 4).
// blockDim = 256 (8 waves); grid = (rows/16, cols/(8*16)).
// ---------------------------------------------------------------------------
template <int K1, int K2, bool RELUA, bool RELUOUT>
__global__ __launch_bounds__(256)
void gemm_wmma_kernel(const float* __restrict__ A, const float* __restrict__ A2,
                      const float* __restrict__ W, const float* __restrict__ bias,
                      float* __restrict__ out, int cols) {
  constexpr int Ktot = K1 + K2;
  const int lane = threadIdx.x & 31;
  const int half = lane >> 4;       // 0: K={0,1}, 1: K={2,3} within each 4-slice
  const int r    = lane & 15;       // A row / B col within the tile
  const int rowBase = blockIdx.x << 4;
  const int colBase = (((int)blockIdx.y << 3) + (threadIdx.x >> 5)) << 4;

  const long long arow = rowBase + r;
  const long long wcol = colBase + r;
  const float* __restrict__ arowp = A + arow * K1;
  const float* __restrict__ wrow  = W + wcol * (long long)Ktot;

  __builtin_prefetch(wrow, 0, 3);
  __builtin_prefetch(arowp, 0, 3);

  v8f acc = {};

  // -------- main loop over K1 (compile-time, guard-free) --------
  const float2* __restrict__ a2p = (const float2*)arowp;   // 8B aligned: K1%2==0
#pragma unroll 4
  for (int k4 = 0; k4 < K1 / 4; ++k4) {
    float2 a = a2p[(k4 << 1) + half];
    if (RELUA) { a.x = fmaxf(a.x, 0.0f); a.y = fmaxf(a.y, 0.0f); }
    float b0, b1;
    if ((Ktot & 1) == 0) {                    // W rows 8B-aligned -> b64 load
      const float2 b = ((const float2*)wrow)[(k4 << 1) + half];
      b0 = b.x; b1 = b.y;
    } else {                                  // odd Ktot (lin_v): scalar loads
      const int ka = (k4 << 2) + (half << 1);
      b0 = wrow[ka];
      b1 = wrow[ka + 1];
    }
    v2f av = {a.x, a.y};
    v2f bv = {b0, b1};
    acc = __builtin_amdgcn_wmma_f32_16x16x4_f32(
        false, av, false, bv, (short)0, acc, false, false);
  }

  // -------- concat tail over K2 (lin_v: 2 iterations, last one guarded) ----
  if (K2 > 0) {
    const float* __restrict__ a2row = A2 + arow * K2;
#pragma unroll
    for (int k = K1; k < K1 + ((K2 + 3) & ~3); k += 4) {
      const int ka = k + (half << 1);
      float a0 = (ka     < Ktot) ? a2row[ka - K1]     : 0.0f;
      float a1 = (ka + 1 < Ktot) ? a2row[ka + 1 - K1] : 0.0f;
      if (RELUA) { a0 = fmaxf(a0, 0.0f); a1 = fmaxf(a1, 0.0f); }
      float b0 = (ka     < Ktot) ? wrow[ka]     : 0.0f;
      float b1 = (ka + 1 < Ktot) ? wrow[ka + 1] : 0.0f;
      v2f av = {a0, a1};
      v2f bv = {b0, b1};
      acc = __builtin_amdgcn_wmma_f32_16x16x4_f32(
          false, av, false, bv, (short)0, acc, false, false);
    }
  }

  // D layout: vgpr p, lane L -> (m = p + 8*(L>>4), n = L&15)
  const float bi = bias ? bias[colBase + r] : 0.0f;
#pragma unroll
  for (int p = 0; p < 8; ++p) {
    const int m = p + (half << 3);
    float v = acc[p] + bi;
    if (RELUOUT) v = fmaxf(v, 0.0f);
    out[(long long)(rowBase + m) * cols + colBase + r] = v;
  }
}

// ---------------------------------------------------------------------------
// Edge MLP: ew[e] = sigmoid( relu(attr@W1^T + b1) @ W2^T + b2 ), 8->8->1
// Weights staged once per block into LDS (81 floats).
// ---------------------------------------------------------------------------
__global__ void edge_mlp_kernel(const float* __restrict__ attr,
                                const float* __restrict__ w1, const float* __restrict__ b1,
                                const float* __restrict__ w2, const float* __restrict__ b2,
                                float* __restrict__ ew, int E) {
  __shared__ float s[81];
  const int t = threadIdx.x;
  if (t < 64)       s[t] = w1[t];
  else if (t < 72)  s[t] = b1[t - 64];
  else if (t < 80)  s[t] = w2[t - 72];
  else if (t == 80) s[t] = b2[0];
  __syncthreads();
  const int e = blockIdx.x * blockDim.x + t;
  if (e >= E) return;
  float a[8];
  {
    const float4 lo = *(const float4*)(attr + (long long)e * 8);
    const float4 hi = *(const float4*)(attr + (long long)e * 8 + 4);
    a[0] = lo.x; a[1] = lo.y; a[2] = lo.z; a[3] = lo.w;
    a[4] = hi.x; a[5] = hi.y; a[6] = hi.z; a[7] = hi.w;
  }
  float acc = s[80];
#pragma unroll
  for (int j = 0; j < 8; ++j) {
    float h = s[64 + j];
#pragma unroll
    for (int k = 0; k < 8; ++k) h = fmaf(s[j * 8 + k], a[k], h);
    acc = fmaf(fmaxf(h, 0.0f), s[72 + j], acc);
  }
  ew[e] = 1.0f / (1.0f + expf(-acc));
}

__global__ void deg_kernel(const float* __restrict__ ew, const int* __restrict__ col,
                           float* __restrict__ deg, int E) {
  int e = blockIdx.x * blockDim.x + threadIdx.x;
  if (e >= E) return;
  atomicAdd(&deg[col[e]], ew[e]);
}

__global__ void dinv_kernel(const float* __restrict__ deg, float* __restrict__ dinv, int n) {
  int i = blockIdx.x * blockDim.x + threadIdx.x;
  if (i >= n) return;
  float d = deg[i];
  dinv[i] = (d > 0.0f) ? rsqrtf(fmaxf(d, 1e-30f)) : 0.0f;
}

__global__ void norm_kernel(const float* __restrict__ ew, const int* __restrict__ row,
                            const int* __restrict__ col, const float* __restrict__ dinv,
                            float* __restrict__ nrm, int E) {
  int e = blockIdx.x * blockDim.x + threadIdx.x;
  if (e >= E) return;
  nrm[e] = dinv[row[e]] * ew[e] * dinv[col[e]];
}

// out[col[e], :] += xw[row[e], :] * norm[e] -- float4 per thread; total = E*32
__global__ void conv_scatter_kernel(const float* __restrict__ xw,
                                    const int* __restrict__ row, const int* __restrict__ col,
                                    const float* __restrict__ nrm,
                                    float* __restrict__ out, long long total) {
  long long gid = (long long)blockIdx.x * blockDim.x + threadIdx.x;
  if (gid >= total) return;
  const int e = (int)(gid >> 5);
  const int t = (int)((gid & 31) << 2);
  const float4 x = *(const float4*)(xw + (long long)row[e] * HID + t);
  const float s = nrm[e];
  float* o = out + (long long)col[e] * HID + t;
  atomicAdd(o + 0, x.x * s);
  atomicAdd(o + 1, x.y * s);
  atomicAdd(o + 2, x.z * s);
  atomicAdd(o + 3, x.w * s);
}

// final: out[n] = dot(X[n, 0:256], w)  (float4 loads)
__global__ void final_dot_kernel(const float* __restrict__ X, const float* __restrict__ w,
                                 float* __restrict__ out, int n) {
  int i = blockIdx.x * blockDim.x + threadIdx.x;
  if (i >= n) return;
  const float4* xr = (const float4*)(X + (long long)i * OEC);
  const float4* wr = (const float4*)w;
  float s = 0.0f;
#pragma unroll 4
  for (int j = 0; j < OEC / 4; ++j) {
    const float4 x = xr[j];
    const float4 ww = wr[j];
    s = fmaf(x.x, ww.x, s);
    s = fmaf(x.y, ww.y, s);
    s = fmaf(x.z, ww.z, s);
    s = fmaf(x.w, ww.w, s);
  }
  out[i] = s;
}

// ---------------------------------------------------------------------------
extern "C" void kernel_launch(void* const* d_in, const int* in_sizes, int n_in,
                              void* d_out, int out_size, void* d_ws, size_t ws_size,
                              hipStream_t stream) {
  const float* e2      = (const float*)d_in[0];
  const int*   idx     = (const int*)  d_in[1];
  const float* afe     = (const float*)d_in[2];
  const int*   bei     = (const int*)  d_in[3];
  const float* battr   = (const float*)d_in[4];
  const float* lin_v_w = (const float*)d_in[5];
  const float* lin_v_b = (const float*)d_in[6];
  const float* conv_w  = (const float*)d_in[7];
  const float* mw1     = (const float*)d_in[8];
  const float* mb1     = (const float*)d_in[9];
  const float* mw2     = (const float*)d_in[10];
  const float* mb2     = (const float*)d_in[11];
  const float* up_w    = (const float*)d_in[12];
  const float* up_b    = (const float*)d_in[13];
  const float* lins_w  = (const float*)d_in[14];
  const float* lins_b  = (const float*)d_in[15];
  const float* lin_w   = (const float*)d_in[16];

  const int M  = in_sizes[0] / HID;   // 600000 message edges
  const int Nn = in_sizes[2] / 7;     // 50000 nodes
  const int E  = in_sizes[3] / 2;     // 1600000 bond edges

  const int* row = bei;
  const int* col = bei + E;

  float* ws   = (float*)d_ws;
  float* bufA = ws;                                   // [N,128]
  float* bufB = bufA + (long long)Nn * HID;           // [N,128]
  float* bufC = bufB + (long long)Nn * HID;           // [N,128]
  float* up   = bufC + (long long)Nn * HID;           // [N,256]
  float* tb   = up   + (long long)Nn * OEC;           // [N,256]
  float* ew   = tb   + (long long)Nn * OEC;           // [E]
  float* nrm  = ew + E;                               // [E]
  float* deg  = nrm + E;                              // [N]
  float* dnv  = deg + Nn;                             // [N]

  dim3 blk(256);
  const int rowTiles = Nn / 16;                       // 3125 exactly

  // 1) v0 = segment_sum(e2, i)
  zero_kernel<<<2048, blk, 0, stream>>>(bufA, (long long)Nn * HID);
  long long tot_e2 = (long long)M * 32;
  scatter_rows_kernel<<<(unsigned)((tot_e2 + 255) / 256), blk, 0, stream>>>(e2, idx, bufA, tot_e2);

  // 2) lin_v: bufB = [v0 || afe] @ lin_v_w^T + b   (K = 128 + 7)
  gemm_wmma_kernel<HID, 7, false, false><<<dim3(rowTiles, 1), blk, 0, stream>>>(
      bufA, afe, lin_v_w, lin_v_b, bufB, HID);

  // 3) edge weights + gcn_norm
  edge_mlp_kernel<<<(E + 255) / 256, blk, 0, stream>>>(battr, mw1, mb1, mw2, mb2, ew, E);
  zero_kernel<<<256, blk, 0, stream>>>(deg, Nn);
  deg_kernel<<<(E + 255) / 256, blk, 0, stream>>>(ew, col, deg, E);
  dinv_kernel<<<(Nn + 255) / 256, blk, 0, stream>>>(deg, dnv, Nn);
  norm_kernel<<<(E + 255) / 256, blk, 0, stream>>>(ew, row, col, dnv, nrm, E);

  // 4) conv #1: xw = v @ conv_w^T ; v = scatter(xw[row]*norm -> col)
  long long tot_ec = (long long)E * 32;
  gemm_wmma_kernel<HID, 0, false, false><<<dim3(rowTiles, 1), blk, 0, stream>>>(
      bufB, nullptr, conv_w, nullptr, bufC, HID);
  zero_kernel<<<2048, blk, 0, stream>>>(bufA, (long long)Nn * HID);
  conv_scatter_kernel<<<(unsigned)((tot_ec + 255) / 256), blk, 0, stream>>>(
      bufC, row, col, nrm, bufA, tot_ec);

  // 5) conv #2
  gemm_wmma_kernel<HID, 0, false, false><<<dim3(rowTiles, 1), blk, 0, stream>>>(
      bufA, nullptr, conv_w, nullptr, bufC, HID);
  zero_kernel<<<2048, blk, 0, stream>>>(bufB, (long long)Nn * HID);
  conv_scatter_kernel<<<(unsigned)((tot_ec + 255) / 256), blk, 0, stream>>>(
      bufC, row, col, nrm, bufB, tot_ec);

  // 6) lin_up with fused input ReLU: up = relu(v) @ up_w^T + b   [N,256]
  gemm_wmma_kernel<HID, 0, true, false><<<dim3(rowTiles, 2), blk, 0, stream>>>(
      bufB, nullptr, up_w, up_b, up, OEC);

  // 7) 3x MLP layers with fused output ReLU (ping-pong up <-> tb)
  gemm_wmma_kernel<OEC, 0, false, true><<<dim3(rowTiles, 2), blk, 0, stream>>>(
      up, nullptr, lins_w + 0LL * OEC * OEC, lins_b + 0 * OEC, tb, OEC);
  gemm_wmma_kernel<OEC, 0, false, true><<<dim3(rowTiles, 2), blk, 0, stream>>>(
      tb, nullptr, lins_w + 1LL * OEC * OEC, lins_b + 1 * OEC, up, OEC);
  gemm_wmma_kernel<OEC, 0, false, true><<<dim3(rowTiles, 2), blk, 0, stream>>>(
      up, nullptr, lins_w + 2LL * OEC * OEC, lins_b + 2 * OEC, tb, OEC);

  // 8) final projection to OC=1
  final_dot_kernel<<<(Nn + 255) / 256, blk, 0, stream>>>(tb, lin_w, (float*)d_out, Nn);
}